// QuanvolutionFilter_65481071407000
// MI455X (gfx1250) — compile-verified
//
#include <hip/hip_runtime.h>
#include <math.h>

typedef __attribute__((ext_vector_type(2))) float v2f;
typedef __attribute__((ext_vector_type(8))) float v8f;

// ---------------------------------------------------------------------------
// Kernel 1: build the fixed 16x16 complex unitary U from params (2,4,3).
// One wave; thread `col` (<16) propagates basis state |col> through the
// circuit, producing column col of U. Stored to workspace:
//   ws[0..255]   = U_re (row-major 16x16)
//   ws[256..511] = U_im
// ---------------------------------------------------------------------------
__device__ __forceinline__ void apply1q(float re[16], float im[16], int mask,
    float g00r, float g00i, float g01r, float g01i,
    float g10r, float g10i, float g11r, float g11i) {
#pragma unroll
  for (int i = 0; i < 16; ++i) {
    if (i & mask) continue;
    int j = i | mask;
    float r0 = re[i], i0 = im[i], r1 = re[j], i1 = im[j];
    re[i] = g00r * r0 - g00i * i0 + g01r * r1 - g01i * i1;
    im[i] = g00r * i0 + g00i * r0 + g01r * i1 + g01i * r1;
    re[j] = g10r * r0 - g10i * i0 + g11r * r1 - g11i * i1;
    im[j] = g10r * i0 + g10i * r0 + g11r * i1 + g11i * r1;
  }
}

__global__ void build_unitary(const float* __restrict__ params,
                              float* __restrict__ U) {
  int col = threadIdx.x;
  if (col >= 16) return;
  float re[16], im[16];
#pragma unroll
  for (int i = 0; i < 16; ++i) { re[i] = 0.0f; im[i] = 0.0f; }
  re[col] = 1.0f;

#pragma unroll
  for (int l = 0; l < 2; ++l) {
#pragma unroll
    for (int w = 0; w < 4; ++w) {
      int mask = 1 << (3 - w);
      float tx = params[(l * 4 + w) * 3 + 0] * 0.5f;
      float ty = params[(l * 4 + w) * 3 + 1] * 0.5f;
      float tz = params[(l * 4 + w) * 3 + 2] * 0.5f;
      float cx = cosf(tx), sx = sinf(tx);
      // RX: [[c, -i s], [-i s, c]]
      apply1q(re, im, mask, cx, 0.0f, 0.0f, -sx, 0.0f, -sx, cx, 0.0f);
      float cy = cosf(ty), sy = sinf(ty);
      // RY: [[c, -s], [s, c]]
      apply1q(re, im, mask, cy, 0.0f, -sy, 0.0f, sy, 0.0f, cy, 0.0f);
      float cz = cosf(tz), sz = sinf(tz);
      // RZ: diag(e^{-i t/2}, e^{+i t/2})
      apply1q(re, im, mask, cz, -sz, 0.0f, 0.0f, 0.0f, 0.0f, cz, sz);
    }
    // CNOT ring: (0,1) (1,2) (2,3) (3,0)
#pragma unroll
    for (int w = 0; w < 4; ++w) {
      int ctrl = (w < 3) ? w : 3;
      int tgt  = (w < 3) ? (w + 1) : 0;
      int mc = 1 << (3 - ctrl), mt = 1 << (3 - tgt);
#pragma unroll
      for (int i = 0; i < 16; ++i) {
        if ((i & mc) && !(i & mt)) {
          int j = i | mt;
          float tr = re[i]; re[i] = re[j]; re[j] = tr;
          float ti = im[i]; im[i] = im[j]; im[j] = ti;
        }
      }
    }
  }
#pragma unroll
  for (int i = 0; i < 16; ++i) {
    U[i * 16 + col]       = re[i];
    U[256 + i * 16 + col] = im[i];
  }
}

// ---------------------------------------------------------------------------
// Kernel 2: one 16-sample tile per wave.
//   V tile (16x16 real encoded amplitudes)  x  U^T  via 4 chained
//   V_WMMA_F32_16X16X4_F32 per re/im accumulator, then |.|^2 and a
//   4-stage signed butterfly (Walsh-Hadamard over the 16 lanes) that
//   yields all four <Z_w> simultaneously at lanes n = 8,4,2,1.
// ---------------------------------------------------------------------------
__global__ void quanv_main(const float* __restrict__ x,
                           const float* __restrict__ U,
                           float* __restrict__ out,
                           int nSamples, int tiles) {
  const int lane = threadIdx.x & 31;
  const int wave = threadIdx.x >> 5;
  const int tile = blockIdx.x * (blockDim.x >> 5) + wave;
  if (tile >= tiles) return;                 // wave-uniform: EXEC stays all-1s
  const int n    = lane & 15;
  const int half = lane >> 4;

  // B operands: B_k[kl][n] = U[n][4k + kl],  kl = r + 2*half (f32 B layout)
  const float* Ure = U;
  const float* Uim = U + 256;
  v2f bre[4], bim[4];
#pragma unroll
  for (int k = 0; k < 4; ++k) {
    int base = n * 16 + 4 * k + 2 * half;
    bre[k][0] = Ure[base];     bre[k][1] = Ure[base + 1];
    bim[k][0] = Uim[base];     bim[k][1] = Uim[base + 1];
  }

  // Per-lane sample (row M = lane%16 of the tile); both halves read the
  // same 4 angles of their sample's 2x2 patch.
  int s = tile * 16 + n;
  if (s >= nSamples) s = nSamples - 1;
  int b  = s / 196;
  int p  = s - b * 196;
  int prow = p / 14;
  int pcol = p - prow * 14;
  const float* xp = x + (b * 784 + prow * 56 + pcol * 2);
  float t0 = xp[0]  * 0.5f;
  float t1 = xp[1]  * 0.5f;
  float t2 = xp[28] * 0.5f;
  float t3 = xp[29] * 0.5f;
  float c0 = cosf(t0), s0 = sinf(t0);
  float c1 = cosf(t1), s1 = sinf(t1);
  float c2 = cosf(t2), s2 = sinf(t2);
  float c3 = cosf(t3), s3 = sinf(t3);

  v8f accre = {0.f, 0.f, 0.f, 0.f, 0.f, 0.f, 0.f, 0.f};
  v8f accim = {0.f, 0.f, 0.f, 0.f, 0.f, 0.f, 0.f, 0.f};
#pragma unroll
  for (int k = 0; k < 4; ++k) {
    // A operand: lane holds v[i] for i = 4k + 2*half + r  (f32 A layout)
    v2f a;
#pragma unroll
    for (int r = 0; r < 2; ++r) {
      int i = 4 * k + 2 * half + r;
      float f0 = (i & 8) ? s0 : c0;   // wire 0 = MSB of basis index
      float f1 = (i & 4) ? s1 : c1;
      float f2 = (i & 2) ? s2 : c2;
      float f3 = (i & 1) ? s3 : c3;
      a[r] = f0 * f1 * f2 * f3;
    }
    accre = __builtin_amdgcn_wmma_f32_16x16x4_f32(
        false, a, false, bre[k], (short)0, accre, false, false);
    accim = __builtin_amdgcn_wmma_f32_16x16x4_f32(
        false, a, false, bim[k], (short)0, accim, false, false);
  }

  // Probabilities: D layout => element (M = r + 8*half, N = lane%16)
  float pz[8];
#pragma unroll
  for (int r = 0; r < 8; ++r)
    pz[r] = accre[r] * accre[r] + accim[r] * accim[r];

  // Walsh-Hadamard over the 16 lanes of each half-wave:
  // lane n ends with H[n] = sum_j (-1)^{popcount(n&j)} P[j].
#pragma unroll
  for (int st = 0; st < 4; ++st) {
    int m = 1 << st;
#pragma unroll
    for (int r = 0; r < 8; ++r) {
      float o = __shfl_xor(pz[r], m, 32);
      pz[r] = (lane & m) ? (o - pz[r]) : (pz[r] + o);
    }
  }

  // <Z_w> = H[1 << (3-w)]  =>  lane 8 -> w0, 4 -> w1, 2 -> w2, 1 -> w3.
  int w = -1;
  if      (n == 8) w = 0;
  else if (n == 4) w = 1;
  else if (n == 2) w = 2;
  else if (n == 1) w = 3;
  if (w >= 0) {
#pragma unroll
    for (int r = 0; r < 8; ++r) {
      int sm = tile * 16 + r + 8 * half;
      if (sm < nSamples) out[sm * 4 + w] = pz[r];  // flat = sample*4 + wire
    }
  }
}

// ---------------------------------------------------------------------------
extern "C" void kernel_launch(void* const* d_in, const int* in_sizes, int n_in,
                              void* d_out, int out_size, void* d_ws, size_t ws_size,
                              hipStream_t stream) {
  const float* x      = (const float*)d_in[0];   // (B,1,28,28) fp32
  const float* params = (const float*)d_in[1];   // (2,4,3) fp32
  float* out = (float*)d_out;                    // (B,784) fp32
  float* U   = (float*)d_ws;                     // 512 floats: U_re | U_im

  int B        = in_sizes[0] / 784;
  int nSamples = B * 196;
  int tiles    = (nSamples + 15) / 16;

  build_unitary<<<1, 32, 0, stream>>>(params, U);

  const int wavesPerBlock = 8;                   // 256 threads = 8 wave32
  int blocks = (tiles + wavesPerBlock - 1) / wavesPerBlock;
  quanv_main<<<blocks, wavesPerBlock * 32, 0, stream>>>(x, U, out, nSamples, tiles);
}